// GlobalPointer_53901839564912
// MI455X (gfx1250) — compile-verified
//
#include <hip/hip_runtime.h>

typedef __attribute__((ext_vector_type(2))) float v2f;
typedef __attribute__((ext_vector_type(8))) float v8f;

#define BATCH   2
#define S_LEN   2048
#define DIM_IN  768
#define NCOL    512      // O*H = 8*64
#define HEADS   8
#define HDIM    64
#define NEGV    1000000000000.0f
// ln(10000)/32  (inv_freq exponent step for head-dim 64 interleaved RoPE)
#define LN1E4_OVER_32 0.28782313662425572f

static __device__ __forceinline__ v8f wmma_f32(v2f a, v2f b, v8f c) {
  // D = A(16x4, f32) * B(4x16, f32) + C(16x16, f32)
  return __builtin_amdgcn_wmma_f32_16x16x4_f32(
      /*neg_a=*/false, a, /*neg_b=*/false, b,
      /*c_mod=*/(short)0, c, /*reuse_a=*/false, /*reuse_b=*/false);
}

// ---------------------------------------------------------------------------
// Kernel A: Y = RoPE(X @ W + bias)
//   X: (4096, 768) row-major, W: (768, 512) row-major, Y: (4096, 512)
// Block = 256 threads (8 waves). Workgroup tile 64(M) x 64(N).
// Wave (wid&3) -> 16-row band; (wid>>2) -> 32-col band (2 accumulators).
// ---------------------------------------------------------------------------
__global__ __launch_bounds__(256) void proj_rope_kernel(
    const float* __restrict__ X, const float* __restrict__ W,
    const float* __restrict__ bias, float* __restrict__ Y) {
  const int lane = threadIdx.x & 31;
  const int wid  = threadIdx.x >> 5;
  const int m0 = blockIdx.x * 64 + (wid & 3) * 16;
  const int n0 = blockIdx.y * 64 + (wid >> 2) * 32;
  const int lr  = lane & 15;        // M index (A) / N index (B,C) within tile
  const int lhk = (lane >> 4) * 2;  // K sub-offset: 0 (lanes 0-15) or 2 (16-31)

  v8f c0 = {}; v8f c1 = {};
  const float* arow = X + (size_t)(m0 + lr) * DIM_IN + lhk;

  for (int k = 0; k < DIM_IN; k += 4) {
    // A frag: {A[m, k+lhk], A[m, k+lhk+1]}  (VGPR0=K0/K2, VGPR1=K1/K3)
    v2f a = *(const v2f*)(arow + k);
    // B frags: B(kk, n) = W[kk*512 + n]; lanes stripe N -> coalesced
    const float* bp = W + (size_t)(k + lhk) * NCOL + n0 + lr;
    v2f b0 = { bp[0],  bp[NCOL] };
    v2f b1 = { bp[16], bp[16 + NCOL] };
    c0 = wmma_f32(a, b0, c0);
    c1 = wmma_f32(a, b1, c1);
  }

  // Epilogue: bias + interleaved RoPE, then store fp32.
  // C layout: VGPR r, lane l -> m = m0 + r + 8*(l>>4), n = nbase + (l&15).
  #pragma unroll
  for (int half = 0; half < 2; ++half) {
    const int n  = n0 + half * 16 + lr;
    const int h  = n & (HDIM - 1);    // position within head
    const int fi = h & 31;            // frequency index (tile-by-2 variant)
    const float bv = bias[n];
    const float inv_freq = __expf(-(float)fi * LN1E4_OVER_32);
    #pragma unroll
    for (int r = 0; r < 8; ++r) {
      const int m = m0 + r + (lane >> 4) * 8;
      const int p = m & (S_LEN - 1);  // sequence position (rows = b*S + p)
      float sn, cs;
      __sincosf((float)p * inv_freq, &sn, &cs);
      float x = (half ? c1[r] : c0[r]) + bv;
      // partner element (m, n^1) sits in lane l^1, same VGPR
      float partner = __shfl_xor(x, 1, 32);
      float rot = (h & 1) ? partner : -partner;  // even: -x[h+1]; odd: +x[h-1]
      Y[(size_t)m * NCOL + n] = x * cs + rot * sn;
    }
  }
}

// ---------------------------------------------------------------------------
// Kernel B: logits[b,o,m,n] = (1/8) * sum_d start[b,m,o,d]*end[b,n,o,d]
//           masked by pad(n) and strict-lower-triangular -NEG.
// NT GEMM, K=64, fully unrolled. Block = 256 threads (8 waves).
// Workgroup tile 64(M) x 128(N); each wave 32x32 (4 accumulators).
// ---------------------------------------------------------------------------
__global__ __launch_bounds__(256) void logits_kernel(
    const float* __restrict__ Sm, const float* __restrict__ Em,
    const float* __restrict__ mask, float* __restrict__ out) {
  const int lane = threadIdx.x & 31;
  const int wid  = threadIdx.x >> 5;
  const int bh = blockIdx.z;              // 0..15 -> (b, head)
  const int b  = bh >> 3;
  const int o  = bh & 7;
  const int m0 = blockIdx.x * 64  + (wid & 1) * 32;
  const int n0 = blockIdx.y * 128 + (wid >> 1) * 32;
  const int lr  = lane & 15;
  const int lhk = (lane >> 4) * 2;

  const float* Sb = Sm + (size_t)b * S_LEN * NCOL + (size_t)o * HDIM;
  const float* Eb = Em + (size_t)b * S_LEN * NCOL + (size_t)o * HDIM;

  const float* a0p = Sb + (size_t)(m0      + lr) * NCOL + lhk;
  const float* a1p = Sb + (size_t)(m0 + 16 + lr) * NCOL + lhk;
  const float* b0p = Eb + (size_t)(n0      + lr) * NCOL + lhk;
  const float* b1p = Eb + (size_t)(n0 + 16 + lr) * NCOL + lhk;

  v8f acc00 = {}, acc01 = {}, acc10 = {}, acc11 = {};
  #pragma unroll
  for (int k = 0; k < HDIM; k += 4) {
    v2f a0 = *(const v2f*)(a0p + k);
    v2f a1 = *(const v2f*)(a1p + k);
    v2f bb0 = *(const v2f*)(b0p + k);   // B(kk,n) = end[n, kk]: contiguous in k
    v2f bb1 = *(const v2f*)(b1p + k);
    acc00 = wmma_f32(a0, bb0, acc00);
    acc01 = wmma_f32(a0, bb1, acc01);
    acc10 = wmma_f32(a1, bb0, acc10);
    acc11 = wmma_f32(a1, bb1, acc11);
  }

  // Epilogue: scale 1/sqrt(64), padding mask (n axis), causal tril (m>n).
  const float scale = 0.125f;
  float* obase = out + (size_t)bh * S_LEN * S_LEN;
  #pragma unroll
  for (int im = 0; im < 2; ++im) {
    #pragma unroll
    for (int in_ = 0; in_ < 2; ++in_) {
      const int nn = n0 + in_ * 16 + lr;
      const float pad = mask[b * S_LEN + nn];
      const float neg_off = (1.0f - pad) * NEGV;
      const v8f& acc = im ? (in_ ? acc11 : acc10) : (in_ ? acc01 : acc00);
      #pragma unroll
      for (int r = 0; r < 8; ++r) {
        const int mm = m0 + im * 16 + r + (lane >> 4) * 8;
        float v = acc[r] * scale;
        v = v * pad - neg_off;
        if (mm > nn) v -= NEGV;
        obase[(size_t)mm * S_LEN + nn] = v;   // lanes stripe n -> coalesced
      }
    }
  }
}

extern "C" void kernel_launch(void* const* d_in, const int* in_sizes, int n_in,
                              void* d_out, int out_size, void* d_ws, size_t ws_size,
                              hipStream_t stream) {
  const float* X    = (const float*)d_in[0];  // (2, 2048, 768)
  const float* mask = (const float*)d_in[1];  // (2, 2048)
  const float* Ws   = (const float*)d_in[2];  // (768, 512)
  const float* bs   = (const float*)d_in[3];  // (512)
  const float* We   = (const float*)d_in[4];  // (768, 512)
  const float* be   = (const float*)d_in[5];  // (512)
  float* out = (float*)d_out;                 // (2, 8, 2048, 2048)

  float* startBuf = (float*)d_ws;                       // 8 MB
  float* endBuf   = startBuf + (size_t)BATCH * S_LEN * NCOL;  // 8 MB

  dim3 blk(256);
  dim3 grdA((BATCH * S_LEN) / 64, NCOL / 64, 1);
  proj_rope_kernel<<<grdA, blk, 0, stream>>>(X, Ws, bs, startBuf);
  proj_rope_kernel<<<grdA, blk, 0, stream>>>(X, We, be, endBuf);

  dim3 grdB(S_LEN / 64, S_LEN / 128, BATCH * HEADS);
  logits_kernel<<<grdB, blk, 0, stream>>>(startBuf, endBuf, mask, out);
}